// Stack_47768626266458
// MI455X (gfx1250) — compile-verified
//
#include <hip/hip_runtime.h>
#include <hip/hip_bf16.h>

#define THREADS 256
#define IPT 16
#define CHUNK (THREADS * IPT)   // 4096 elements of s per block
#define COPY_IPT 8              // float4s per thread in copy kernel (32KB LDS stage)

#define AS1 __attribute__((address_space(1)))
#define AS3 __attribute__((address_space(3)))

typedef int v4i_t __attribute__((vector_size(16)));

#if defined(__gfx1250__) && \
    __has_builtin(__builtin_amdgcn_global_load_async_to_lds_b128) && \
    __has_builtin(__builtin_amdgcn_global_store_async_from_lds_b128)
#define USE_ASYNC_COPY 1
#else
#define USE_ASYNC_COPY 0
#endif

__device__ __forceinline__ void wait_asynccnt0() {
#if __has_builtin(__builtin_amdgcn_s_wait_asynccnt)
    __builtin_amdgcn_s_wait_asynccnt(0);
#else
    asm volatile("s_wait_asynccnt 0" ::: "memory");
#endif
}

// ---------------------------------------------------------------------------
// Copy V (N*E floats) -> V2 via the CDNA5 async global<->LDS path.
// Fast path: full 2048-float4 tile, uniform branch, no per-op EXEC juggling;
// all addresses are base + compile-time-constant offsets.
// ---------------------------------------------------------------------------
__global__ void copy_v_kernel(const float4* __restrict__ src,
                              float4* __restrict__ dst, long long n4) {
    const long long tileBase = (long long)blockIdx.x * (THREADS * COPY_IPT);
#if USE_ASYNC_COPY
    __shared__ float4 stage[THREADS * COPY_IPT];
    if (tileBase + (THREADS * COPY_IPT) <= n4) {       // uniform over the block
        const float4* sp = src + tileBase + threadIdx.x;
        float4*       dp = dst + tileBase + threadIdx.x;
#pragma unroll
        for (int k = 0; k < COPY_IPT; ++k) {
            __builtin_amdgcn_global_load_async_to_lds_b128(
                (AS1 v4i_t*)(sp + k * THREADS),
                (AS3 v4i_t*)&stage[k * THREADS + threadIdx.x], 0, 0);
        }
        wait_asynccnt0();
#pragma unroll
        for (int k = 0; k < COPY_IPT; ++k) {
            __builtin_amdgcn_global_store_async_from_lds_b128(
                (AS1 v4i_t*)(dp + k * THREADS),
                (AS3 v4i_t*)&stage[k * THREADS + threadIdx.x], 0, 0);
        }
        // s_endpgm performs an implicit wait-idle; stores complete before exit.
    } else {
#pragma unroll
        for (int k = 0; k < COPY_IPT; ++k) {
            long long i = tileBase + (long long)k * THREADS + threadIdx.x;
            if (i < n4) dst[i] = src[i];
        }
    }
#else
#pragma unroll
    for (int k = 0; k < COPY_IPT; ++k) {
        long long i = tileBase + (long long)k * THREADS + threadIdx.x;
        if (i < n4) dst[i] = src[i];
    }
#endif
}

// ---------------------------------------------------------------------------
// Per-block sums of s (block b covers [b*CHUNK, (b+1)*CHUNK)).
// ---------------------------------------------------------------------------
__global__ void block_sums_kernel(const float* __restrict__ s,
                                  float* __restrict__ blockSums, long long N) {
    const int t = threadIdx.x;
    const long long start = (long long)blockIdx.x * CHUNK;
    float acc = 0.f;
#pragma unroll
    for (int k = 0; k < IPT; ++k) {
        long long i = start + (long long)k * THREADS + t;
        if (i < N) acc += s[i];
    }
    __shared__ float sc[THREADS];
    sc[t] = acc;
    __syncthreads();
    for (int off = THREADS >> 1; off > 0; off >>= 1) {
        if (t < off) sc[t] += sc[t + off];
        __syncthreads();
    }
    if (t == 0) blockSums[blockIdx.x] = sc[0];
}

// ---------------------------------------------------------------------------
// Single-block reverse exclusive scan: out[b] = sum_{b' > b} in[b'], out[B] = total.
// Implemented as a forward scan over the reversed sequence with a carry.
// ---------------------------------------------------------------------------
__global__ void reverse_scan_kernel(const float* __restrict__ in,
                                    float* __restrict__ out, int B) {
    __shared__ float tile[1024];
    __shared__ float carry_s;
    const int t = threadIdx.x;
    if (t == 0) carry_s = 0.f;
    __syncthreads();
    const int nt = (B + 1023) / 1024;
    for (int ti = 0; ti < nt; ++ti) {
        int j = ti * 1024 + t;      // position in reversed order
        int i = B - 1 - j;          // original index
        float x = (j < B) ? in[i] : 0.f;
        tile[t] = x;
        __syncthreads();
        for (int off = 1; off < 1024; off <<= 1) {
            float y = tile[t];
            if (t >= off) y += tile[t - off];
            __syncthreads();
            tile[t] = y;
            __syncthreads();
        }
        float incl = tile[t];
        float carry = carry_s;
        if (j < B) out[i] = carry + (incl - x);   // exclusive suffix
        __syncthreads();
        if (t == 1023) carry_s = carry + tile[1023];
        __syncthreads();
    }
    if (t == 0) out[B] = carry_s;   // total
}

// Shared helper: given per-thread total, return exclusive suffix over threads
// with higher t in the block (computed via reversed in-block scan, no large
// cancellation). sc must be THREADS floats of LDS; leaves sc dirty.
__device__ __forceinline__ float thread_suffix_excl(float threadTotal, float* sc) {
    const int t = threadIdx.x;
    sc[THREADS - 1 - t] = threadTotal;
    __syncthreads();
    for (int off = 1; off < THREADS; off <<= 1) {
        float y = sc[t];
        if (t >= off) y += sc[t - off];
        __syncthreads();
        sc[t] = y;
        __syncthreads();
    }
    return sc[THREADS - 1 - t] - threadTotal;
}

// ---------------------------------------------------------------------------
// Pass 2: s_new[i] = relu(s[i] - relu(u - suf[i]));  write s2[i]; per-block
// sums of s_new.
// ---------------------------------------------------------------------------
__global__ void strength_update_kernel(const float* __restrict__ s,
                                       const float* __restrict__ uPtr,
                                       const float* __restrict__ blockSufS,
                                       float* __restrict__ s2,
                                       float* __restrict__ blockSums2,
                                       long long N) {
    const int t = threadIdx.x;
    const long long base = (long long)blockIdx.x * CHUNK + (long long)t * IPT;
    float x[IPT];
    if (base + IPT <= N) {
        const float4* p = (const float4*)(s + base);
#pragma unroll
        for (int q = 0; q < IPT / 4; ++q) {
            float4 v4 = p[q];
            x[4 * q + 0] = v4.x; x[4 * q + 1] = v4.y;
            x[4 * q + 2] = v4.z; x[4 * q + 3] = v4.w;
        }
    } else {
#pragma unroll
        for (int k = 0; k < IPT; ++k) {
            long long i = base + k;
            x[k] = (i < N) ? s[i] : 0.f;
        }
    }
    // thread-local exclusive suffix, accumulated in reverse (accurate tail)
    float sufLocal[IPT];
    float run = 0.f;
#pragma unroll
    for (int k = IPT - 1; k >= 0; --k) { sufLocal[k] = run; run += x[k]; }

    __shared__ float sc[THREADS];
    float threadSuf = thread_suffix_excl(run, sc);
    const float blockSuf = blockSufS[blockIdx.x];
    const float u = uPtr[0];

    float y[IPT];
    float sumNew = 0.f;
#pragma unroll
    for (int k = 0; k < IPT; ++k) {
        float suf = blockSuf + threadSuf + sufLocal[k];
        float sn = fmaxf(x[k] - fmaxf(u - suf, 0.f), 0.f);
        y[k] = sn;
        sumNew += sn;     // out-of-range lanes contribute exactly 0
    }
    if (base + IPT <= N) {
        float4* p = (float4*)(s2 + base);
#pragma unroll
        for (int q = 0; q < IPT / 4; ++q)
            p[q] = make_float4(y[4 * q + 0], y[4 * q + 1], y[4 * q + 2], y[4 * q + 3]);
    } else {
#pragma unroll
        for (int k = 0; k < IPT; ++k) {
            long long i = base + k;
            if (i < N) s2[i] = y[k];
        }
    }
    __syncthreads();               // sc reuse
    sc[t] = sumNew;
    __syncthreads();
    for (int off = THREADS >> 1; off > 0; off >>= 1) {
        if (t < off) sc[t] += sc[t + off];
        __syncthreads();
    }
    if (t == 0) blockSums2[blockIdx.x] = sc[0];
}

// ---------------------------------------------------------------------------
// Pass 3: per-block partial of r = sum_i min(s_new[i], relu(suf2[i] + d)).
// ---------------------------------------------------------------------------
__global__ void read_partial_kernel(const float* __restrict__ sNew,
                                    const float* __restrict__ dPtr,
                                    const float* __restrict__ blockSuf2,
                                    float* __restrict__ partialR,
                                    long long N) {
    const int t = threadIdx.x;
    const long long base = (long long)blockIdx.x * CHUNK + (long long)t * IPT;
    float x[IPT];
    if (base + IPT <= N) {
        const float4* p = (const float4*)(sNew + base);
#pragma unroll
        for (int q = 0; q < IPT / 4; ++q) {
            float4 v4 = p[q];
            x[4 * q + 0] = v4.x; x[4 * q + 1] = v4.y;
            x[4 * q + 2] = v4.z; x[4 * q + 3] = v4.w;
        }
    } else {
#pragma unroll
        for (int k = 0; k < IPT; ++k) {
            long long i = base + k;
            x[k] = (i < N) ? sNew[i] : 0.f;
        }
    }
    float sufLocal[IPT];
    float run = 0.f;
#pragma unroll
    for (int k = IPT - 1; k >= 0; --k) { sufLocal[k] = run; run += x[k]; }

    __shared__ float sc[THREADS];
    float threadSuf = thread_suffix_excl(run, sc);
    const float blockSuf = blockSuf2[blockIdx.x];
    const float dv = dPtr[0];

    float acc = 0.f;
#pragma unroll
    for (int k = 0; k < IPT; ++k) {
        float suf2 = blockSuf + threadSuf + sufLocal[k] + dv;
        acc += fminf(x[k], fmaxf(suf2, 0.f));  // OOB lanes: min(0, >=0) == 0
    }
    __syncthreads();
    sc[t] = acc;
    __syncthreads();
    for (int off = THREADS >> 1; off > 0; off >>= 1) {
        if (t < off) sc[t] += sc[t + off];
        __syncthreads();
    }
    if (t == 0) partialR[blockIdx.x] = sc[0];
}

// ---------------------------------------------------------------------------
// Finalize: r = sum(partialR) + min(d, relu(0)); append v row; s2[N] = d.
// ---------------------------------------------------------------------------
__global__ void finalize_kernel(const float* __restrict__ partialR, int B,
                                const float* __restrict__ vvec, int E,
                                const float* __restrict__ dPtr,
                                float* __restrict__ out, long long N) {
    const int t = threadIdx.x;
    __shared__ float sc[THREADS];
    float acc = 0.f;
    for (int i = t; i < B; i += THREADS) acc += partialR[i];
    sc[t] = acc;
    __syncthreads();
    for (int off = THREADS >> 1; off > 0; off >>= 1) {
        if (t < off) sc[t] += sc[t + off];
        __syncthreads();
    }
    const float dv = dPtr[0];
    if (t == 0) out[0] = sc[0] + fminf(dv, 0.f);   // last element: min(d, relu(0))
    float* V2 = out + 1;
    if (t < E) V2[N * (long long)E + t] = vvec[t]; // appended pushed vector
    float* s2 = out + 1 + (N + 1) * (long long)E;
    if (t == 0) s2[N] = dv;                        // appended push strength
}

extern "C" void kernel_launch(void* const* d_in, const int* in_sizes, int n_in,
                              void* d_out, int out_size, void* d_ws, size_t ws_size,
                              hipStream_t stream) {
    const float* V = (const float*)d_in[0];
    const float* s = (const float*)d_in[1];
    const float* v = (const float*)d_in[2];
    const float* u = (const float*)d_in[3];
    const float* d = (const float*)d_in[4];

    const long long N = (long long)in_sizes[1];
    const int E = in_sizes[2];

    float* out = (float*)d_out;
    float* V2  = out + 1;
    float* s2  = out + 1 + (N + 1) * (long long)E;

    const int B = (int)((N + CHUNK - 1) / CHUNK);
    float* ws = (float*)d_ws;
    float* blockSumsS = ws;               // B
    float* blockSufS  = ws + B;           // B+1 (index B = total)
    float* blockSums2 = ws + 2 * B + 1;   // B
    float* blockSuf2  = ws + 3 * B + 1;   // B+1 (index B = total)
    float* partialR   = ws + 4 * B + 2;   // B

    // Bulk V -> V2 copy (bandwidth-dominant), async global<->LDS path.
    const long long n4 = (N * (long long)E) / 4;
    const int copyBlocks = (int)((n4 + (THREADS * COPY_IPT - 1)) / (THREADS * COPY_IPT));
    copy_v_kernel<<<copyBlocks, THREADS, 0, stream>>>((const float4*)V, (float4*)V2, n4);

    block_sums_kernel<<<B, THREADS, 0, stream>>>(s, blockSumsS, N);
    reverse_scan_kernel<<<1, 1024, 0, stream>>>(blockSumsS, blockSufS, B);
    strength_update_kernel<<<B, THREADS, 0, stream>>>(s, u, blockSufS, s2, blockSums2, N);
    reverse_scan_kernel<<<1, 1024, 0, stream>>>(blockSums2, blockSuf2, B);
    read_partial_kernel<<<B, THREADS, 0, stream>>>(s2, d, blockSuf2, partialR, N);
    finalize_kernel<<<1, THREADS, 0, stream>>>(partialR, B, v, E, d, out, N);
}